// CBowNEGModel_75153337745588
// MI455X (gfx1250) — compile-verified
//
#include <hip/hip_runtime.h>

// CBOW negative-sampling loss for MI455X (gfx1250, wave32).
// Memory-bound gather problem (~13.4 MB traffic, L2-resident tables);
// dot products done with native-f32 V_WMMA_F32_16X16X4_F32 (diag of 16x16 tile).

typedef __attribute__((ext_vector_type(2))) float v2f;
typedef __attribute__((ext_vector_type(4))) float v4f;
typedef __attribute__((ext_vector_type(8))) float v8f;

#define D_DIM 128
#define C_CTX 10
#define K_NEG 5
#define ROW_PAD 132          // 128 + 4 floats: bank-conflict-free fragment reads
#define WAVES_PER_BLOCK 2    // 2 waves * 16 batches = 32 batches per block

__device__ __forceinline__ float log_sigmoid(float x) {
    // stable: min(x,0) - log1p(exp(-|x|))
    return fminf(x, 0.0f) - log1pf(expf(-fabsf(x)));
}

__global__ __launch_bounds__(32 * WAVES_PER_BLOCK)
void cbow_neg_kernel(const int* __restrict__ pos_u,
                     const int* __restrict__ pos_w,
                     const int* __restrict__ neg_w,
                     const float* __restrict__ u_emb,
                     const float* __restrict__ w_emb,
                     float* __restrict__ block_part)
{
    __shared__ float u_tile[WAVES_PER_BLOCK][16 * ROW_PAD];
    __shared__ float w_tile[WAVES_PER_BLOCK][16 * ROW_PAD];
    __shared__ float wave_sum[WAVES_PER_BLOCK];

    const int lane = threadIdx.x & 31;
    const int wv   = threadIdx.x >> 5;
    const int b0   = (blockIdx.x * WAVES_PER_BLOCK + wv) * 16;

    // ---- Phase A: u_embed = sum_c u_emb[pos_u[b,c]]  ->  LDS tile (16 x 128)
    // Each row load: 32 lanes x float4 = 512B contiguous (global_load_b128).
    for (int r = 0; r < 16; ++r) {
        const int* pu = pos_u + (size_t)(b0 + r) * C_CTX;
        v4f acc = {0.f, 0.f, 0.f, 0.f};
        #pragma unroll
        for (int c = 0; c < C_CTX; ++c) {
            const int idx = pu[c];  // wave-uniform -> scalar load
            const v4f* row = (const v4f*)(u_emb + (size_t)idx * D_DIM);
            acc += row[lane];
        }
        *(v4f*)&u_tile[wv][r * ROW_PAD + lane * 4] = acc;
    }

    // A and B fragments share the same LDS addressing pattern:
    // lane l holds (row = l%16, cols = 4*kc + 2*(l/16) + {0,1}).
    const int   fr   = lane & 15;
    const int   half = lane >> 4;
    const float* ur  = &u_tile[wv][fr * ROW_PAD];
    const float* wr  = &w_tile[wv][fr * ROW_PAD];

    float part = 0.0f;

    for (int s = 0; s < 1 + K_NEG; ++s) {
        __syncthreads();
        // ---- Phase B: stage this slot's 16 w rows (coalesced) ----
        for (int r = 0; r < 16; ++r) {
            const int widx = (s == 0) ? pos_w[b0 + r]
                                      : neg_w[(size_t)(b0 + r) * K_NEG + (s - 1)];
            const v4f* row = (const v4f*)(w_emb + (size_t)widx * D_DIM);
            *(v4f*)&w_tile[wv][r * ROW_PAD + lane * 4] = row[lane];
        }
        __syncthreads();

        // ---- Phase C: 16x16 tile of dots = U(16x128) * Ws^T via chained
        //      V_WMMA_F32_16X16X4_F32 (native f32, K=4 per step, 32 steps) ----
        v8f acc = {0.f, 0.f, 0.f, 0.f, 0.f, 0.f, 0.f, 0.f};
        #pragma unroll 8
        for (int kc = 0; kc < D_DIM / 4; ++kc) {
            const int col = kc * 4 + 2 * half;
            v2f a = *(const v2f*)(ur + col);   // ds_load_b64, bank-conflict-free
            v2f b = *(const v2f*)(wr + col);
            acc = __builtin_amdgcn_wmma_f32_16x16x4_f32(
                false, a, false, b, (short)0, acc, false, false);
        }

        // ---- Phase D: pull the diagonal.
        // D layout: lanes 0-15 hold (M=v, N=lane); lanes 16-31 hold (M=v+8, N=lane-16).
        // diag i<8  -> lane i,    vgpr i
        // diag i>=8 -> lane i+16, vgpr i-8
        const int sel = (lane < 8) ? lane : ((lane >= 24) ? (lane - 24) : -1);
        if (sel >= 0) {
            float dot;
            switch (sel) {
                case 0: dot = acc[0]; break;
                case 1: dot = acc[1]; break;
                case 2: dot = acc[2]; break;
                case 3: dot = acc[3]; break;
                case 4: dot = acc[4]; break;
                case 5: dot = acc[5]; break;
                case 6: dot = acc[6]; break;
                default: dot = acc[7]; break;
            }
            part += log_sigmoid((s == 0) ? dot : -dot);
        }
    }

    // ---- wave tree reduction (fixed order -> deterministic) ----
    #pragma unroll
    for (int off = 16; off > 0; off >>= 1)
        part += __shfl_xor(part, off, 32);
    if (lane == 0) wave_sum[wv] = part;
    __syncthreads();
    if (threadIdx.x == 0) {
        float t = 0.f;
        #pragma unroll
        for (int i = 0; i < WAVES_PER_BLOCK; ++i) t += wave_sum[i];
        block_part[blockIdx.x] = t;
    }
}

// Deterministic final reduction: single wave, fixed strided order + xor tree.
__global__ __launch_bounds__(32)
void cbow_reduce_kernel(const float* __restrict__ parts, int n, float* __restrict__ out)
{
    const int lane = threadIdx.x;
    float s = 0.f;
    for (int i = lane; i < n; i += 32) s += parts[i];
    #pragma unroll
    for (int off = 16; off > 0; off >>= 1)
        s += __shfl_xor(s, off, 32);
    if (lane == 0) out[0] = -s;   // loss = -(sum log-sigmoids)
}

extern "C" void kernel_launch(void* const* d_in, const int* in_sizes, int n_in,
                              void* d_out, int out_size, void* d_ws, size_t ws_size,
                              hipStream_t stream)
{
    const int*   pos_u = (const int*)d_in[0];     // [B, C]
    const int*   pos_w = (const int*)d_in[1];     // [B]
    const int*   neg_w = (const int*)d_in[2];     // [B, K]
    const float* u_emb = (const float*)d_in[3];   // [V, D]
    const float* w_emb = (const float*)d_in[4];   // [V, D]
    float*       out   = (float*)d_out;

    const int B      = in_sizes[1];               // 16384
    const int groups = B / 16;                    // 16 batches per wave
    const int blocks = groups / WAVES_PER_BLOCK;  // 512

    float* parts = (float*)d_ws;                  // blocks * 4 bytes scratch

    cbow_neg_kernel<<<blocks, 32 * WAVES_PER_BLOCK, 0, stream>>>(
        pos_u, pos_w, neg_w, u_emb, w_emb, parts);
    cbow_reduce_kernel<<<1, 32, 0, stream>>>(parts, blocks, out);
}